// NodeGCN_62328565399617
// MI455X (gfx1250) — compile-verified
//
#include <hip/hip_runtime.h>
#include <hip/hip_bf16.h>

typedef float v2f __attribute__((ext_vector_type(2)));
typedef float v8f __attribute__((ext_vector_type(8)));

// ---------------------------------------------------------------------------
// CDNA5 async global->LDS B128 copy (tracked by ASYNCcnt, bypasses VGPRs)
// ---------------------------------------------------------------------------
__device__ __forceinline__ void async_copy_b128(unsigned lds_byte_off, const float* gptr) {
    asm volatile("global_load_async_to_lds_b128 %0, %1, off"
                 :: "v"(lds_byte_off), "v"(gptr)
                 : "memory");
}

__device__ __forceinline__ void wait_async_all() {
    asm volatile("s_wait_asynccnt 0x0" ::: "memory");
}

// ---------------------------------------------------------------------------
// Degree / normalization kernels
// ---------------------------------------------------------------------------
__global__ void k_init_deg(float* __restrict__ deg, int n) {
    int i = blockIdx.x * 256 + threadIdx.x;
    if (i < n) deg[i] = 1.0f;               // self-loop contributes 1 to degree
}

__global__ void k_count_deg(const int* __restrict__ col, float* __restrict__ deg, int E) {
    int e = blockIdx.x * 256 + threadIdx.x;
    if (e < E) atomicAdd(&deg[col[e]], 1.0f);
}

__global__ void k_rsqrt_inplace(float* __restrict__ d, int n) {
    int i = blockIdx.x * 256 + threadIdx.x;
    if (i < n) d[i] = rsqrtf(d[i]);         // deg >= 1 always (self loops)
}

// ---------------------------------------------------------------------------
// Tiled f32 GEMM with CDNA5 WMMA + double-buffered async LDS staging.
// C[M,N] = A[M,K] * W[K,N]. Block = 256 threads (8 waves). Tile 64x64, BK=32.
// Wave w owns two 16x16 C tiles: rows (w%4)*16, cols (w/4)*32 and +16.
// A frag per ISA layout: a[v] = A[lane%16][kk + v + 2*(lane/16)]
// B frag per ISA layout: b[v] = W[kk + v + 2*(lane/16)][lane%16]
// ---------------------------------------------------------------------------
__global__ __launch_bounds__(256) void gcn_gemm_f32_wmma(
    const float* __restrict__ A, const float* __restrict__ W,
    float* __restrict__ C, int M, int K, int N)
{
    constexpr int BM = 64, BN = 64, BK = 32;
    constexpr int LDA = BK + 4;   // 36 floats = 144B row stride (16B multiple)
    constexpr int LDB = BN + 16;  // 80 floats = 320B row stride (16B multiple)
    __shared__ __align__(16) float As[2][BM * LDA];
    __shared__ __align__(16) float Bs[2][BK * LDB];

    const int mbase = blockIdx.y * BM;
    const int nbase = blockIdx.x * BN;
    const int tid   = threadIdx.x;
    const int lane  = tid & 31;
    const int wave  = tid >> 5;
    const int lidx  = lane & 15;
    const int lhalf = lane >> 4;
    const int wr    = (wave & 3) * 16;   // wave row offset in tile
    const int wc    = (wave >> 2) * 32;  // wave col offset in tile

    // Issue one K-tile's async loads into LDS buffer `buf`.
    // A tile: 64 rows x 8 16B-chunks = 512 chunks; B tile: 32 rows x 16 chunks.
    auto stage = [&](int buf, int k0) {
#pragma unroll
        for (int j = 0; j < 2; ++j) {
            int id = tid + j * 256;
            int r  = id >> 3;
            int c4 = (id & 7) << 2;
            int gm = mbase + r;
            unsigned lds = (unsigned)(unsigned long long)(const void*)&As[buf][r * LDA + c4];
            if (gm < M)   // EXEC-masks OOB lanes; stale rows land only in discarded C rows
                async_copy_b128(lds, A + (long)gm * K + k0 + c4);
        }
#pragma unroll
        for (int j = 0; j < 2; ++j) {
            int id = tid + j * 256;
            int r  = id >> 4;
            int c4 = (id & 15) << 2;
            unsigned lds = (unsigned)(unsigned long long)(const void*)&Bs[buf][r * LDB + c4];
            async_copy_b128(lds, W + (long)(k0 + r) * N + nbase + c4);
        }
    };

    v8f c0 = {};
    v8f c1 = {};

    const int T = K / BK;
    stage(0, 0);

    for (int t = 0; t < T; ++t) {
        wait_async_all();       // this wave's async writes to LDS have landed
        __syncthreads();        // everyone's landed AND everyone done reading buf being reused
        if (t + 1 < T) stage((t + 1) & 1, (t + 1) * BK);   // overlap next fetch with compute

        const float* Ab = As[t & 1];
        const float* Bb = Bs[t & 1];
#pragma unroll
        for (int kk = 0; kk < BK; kk += 4) {
            v2f a, b0, b1;
            const float* ap = &Ab[(wr + lidx) * LDA + kk + 2 * lhalf];
            a.x = ap[0];
            a.y = ap[1];
            const int brow = kk + 2 * lhalf;
            b0.x = Bb[brow * LDB + wc + lidx];
            b0.y = Bb[(brow + 1) * LDB + wc + lidx];
            b1.x = Bb[brow * LDB + wc + 16 + lidx];
            b1.y = Bb[(brow + 1) * LDB + wc + 16 + lidx];
            // (neg_a, A, neg_b, B, c_mod, C, reuse_a, reuse_b)
            c0 = __builtin_amdgcn_wmma_f32_16x16x4_f32(false, a, false, b0,
                                                       (short)0, c0, false, false);
            c1 = __builtin_amdgcn_wmma_f32_16x16x4_f32(false, a, false, b1,
                                                       (short)0, c1, false, false);
        }
    }

    // Store: C layout row = v + 8*(lane/16), col = lane%16
#pragma unroll
    for (int v = 0; v < 8; ++v) {
        int gm = mbase + wr + v + 8 * lhalf;
        if (gm < M) {
            C[(long)gm * N + nbase + wc + lidx]      = c0[v];
            C[(long)gm * N + nbase + wc + 16 + lidx] = c1[v];
        }
    }
}

// ---------------------------------------------------------------------------
// Aggregation init: agg[i,:] = h[i,:] * dinv[i]^2 (self-loop term + zero init)
// float4-vectorized; D4 = D/4 is a power of two.
// ---------------------------------------------------------------------------
__global__ void k_self_init(const float4* __restrict__ h, const float* __restrict__ dinv,
                            float4* __restrict__ agg, int n, int D4)
{
    long i = (long)blockIdx.x * 256 + threadIdx.x;
    long total = (long)n * D4;
    if (i < total) {
        int node = (int)(i / D4);
        float w = dinv[node];
        float s = w * w;
        float4 v = h[i];
        v.x *= s; v.y *= s; v.z *= s; v.w *= s;
        agg[i] = v;
    }
}

// ---------------------------------------------------------------------------
// Edge scatter: one wave per edge, lanes stride channels (coalesced atomics).
// agg[col,:] += h[row,:] * dinv[row]*dinv[col]
// ---------------------------------------------------------------------------
__global__ __launch_bounds__(256) void k_edge_scatter(
    const int* __restrict__ rows, const int* __restrict__ cols,
    const float* __restrict__ dinv, const float* __restrict__ h,
    float* __restrict__ agg, int E, int D)
{
    int e = blockIdx.x * 8 + (threadIdx.x >> 5);
    if (e >= E) return;
    int lane = threadIdx.x & 31;
    int r = rows[e];
    int c = cols[e];
    float w = dinv[r] * dinv[c];
    const float* hs = h + (long)r * D;
    float* ad = agg + (long)c * D;
    for (int j = lane; j < D; j += 32)
        atomicAdd(&ad[j], hs[j] * w);
}

// ---------------------------------------------------------------------------
// Finalize: x = relu(x + b) (or no relu for the last layer); float4-vectorized
// ---------------------------------------------------------------------------
__global__ void k_finalize(float4* __restrict__ agg, const float4* __restrict__ b,
                           int n, int D4, int do_relu)
{
    long i = (long)blockIdx.x * 256 + threadIdx.x;
    long total = (long)n * D4;
    if (i < total) {
        float4 bb = b[(int)(i & (D4 - 1))];
        float4 v = agg[i];
        v.x += bb.x; v.y += bb.y; v.z += bb.z; v.w += bb.w;
        if (do_relu) {
            v.x = fmaxf(v.x, 0.0f); v.y = fmaxf(v.y, 0.0f);
            v.z = fmaxf(v.z, 0.0f); v.w = fmaxf(v.w, 0.0f);
        }
        agg[i] = v;
    }
}

// ---------------------------------------------------------------------------
// Launch
// ---------------------------------------------------------------------------
extern "C" void kernel_launch(void* const* d_in, const int* in_sizes, int n_in,
                              void* d_out, int out_size, void* d_ws, size_t ws_size,
                              hipStream_t stream) {
    const float* x  = (const float*)d_in[0];
    const int*   ei = (const int*)d_in[1];     // edge_index flattened [2, E] (int32 per harness)
    const float* W1 = (const float*)d_in[2];
    const float* b1 = (const float*)d_in[3];
    const float* W2 = (const float*)d_in[4];
    const float* b2 = (const float*)d_in[5];
    const float* W3 = (const float*)d_in[6];
    const float* b3 = (const float*)d_in[7];
    float* out = (float*)d_out;

    const int Nn = in_sizes[0] / 512;          // 100000 nodes
    const int E  = in_sizes[1] / 2;            // 1600000 edges
    const int* rows = ei;                      // edge_index[0] = sources (gather)
    const int* cols = ei + E;                  // edge_index[1] = targets (scatter)

    // Workspace layout: dinv[Nn] | bufA[Nn*256] | bufB[Nn*256]
    char* ws = (char*)d_ws;
    size_t dinv_bytes = ((size_t)Nn * sizeof(float) + 255) & ~(size_t)255;
    float* dinv = (float*)ws;
    float* bufA = (float*)(ws + dinv_bytes);
    float* bufB = bufA + (size_t)Nn * 256;

    dim3 blk(256);
    int nb_nodes = (Nn + 255) / 256;
    int nb_edges = (E + 255) / 256;

    // deg -> dinv
    k_init_deg<<<nb_nodes, blk, 0, stream>>>(dinv, Nn);
    k_count_deg<<<nb_edges, blk, 0, stream>>>(cols, dinv, E);
    k_rsqrt_inplace<<<nb_nodes, blk, 0, stream>>>(dinv, Nn);

    dim3 gemm_grid_256(256 / 64, (Nn + 63) / 64);
    dim3 gemm_grid_64(64 / 64, (Nn + 63) / 64);
    int nb_vec256 = (int)(((long)Nn * 64 + 255) / 256);   // Nn*256/4 elements
    int nb_vec64  = (int)(((long)Nn * 16 + 255) / 256);   // Nn*64/4 elements
    int nb_scatter = (E + 7) / 8;

    // ---- Layer 1: 512 -> 256, ReLU ----
    gcn_gemm_f32_wmma<<<gemm_grid_256, blk, 0, stream>>>(x, W1, bufA, Nn, 512, 256);
    k_self_init<<<nb_vec256, blk, 0, stream>>>((const float4*)bufA, dinv, (float4*)bufB, Nn, 64);
    k_edge_scatter<<<nb_scatter, blk, 0, stream>>>(rows, cols, dinv, bufA, bufB, E, 256);
    k_finalize<<<nb_vec256, blk, 0, stream>>>((float4*)bufB, (const float4*)b1, Nn, 64, 1);

    // ---- Layer 2: 256 -> 256, ReLU ----
    gcn_gemm_f32_wmma<<<gemm_grid_256, blk, 0, stream>>>(bufB, W2, bufA, Nn, 256, 256);
    k_self_init<<<nb_vec256, blk, 0, stream>>>((const float4*)bufA, dinv, (float4*)bufB, Nn, 64);
    k_edge_scatter<<<nb_scatter, blk, 0, stream>>>(rows, cols, dinv, bufA, bufB, E, 256);
    k_finalize<<<nb_vec256, blk, 0, stream>>>((float4*)bufB, (const float4*)b2, Nn, 64, 1);

    // ---- Layer 3: 256 -> 64, no ReLU, straight into d_out ----
    gcn_gemm_f32_wmma<<<gemm_grid_64, blk, 0, stream>>>(bufB, W3, bufA, Nn, 256, 64);
    k_self_init<<<nb_vec64, blk, 0, stream>>>((const float4*)bufA, dinv, (float4*)out, Nn, 16);
    k_edge_scatter<<<nb_scatter, blk, 0, stream>>>(rows, cols, dinv, bufA, out, E, 64);
    k_finalize<<<nb_vec64, blk, 0, stream>>>((float4*)out, (const float4*)b3, Nn, 16, 0);
}